// ContentLoss_75436805587632
// MI455X (gfx1250) — compile-verified
//
#include <hip/hip_runtime.h>

typedef __attribute__((ext_vector_type(2))) float v2f;
typedef __attribute__((ext_vector_type(8))) float v8f;

#define WID   61
#define PAD   30
#define HALO  60            // WID-1
#define W     512
#define H     512
#define HW    (512*512)
#define NIMG  16
#define KSTEPS 19           // K = 76 = 19 * 4
#define XCHUNK 128          // columns per hconv wave
#define YCHUNK 128          // rows per vconv wave
#define LDW   (XCHUNK + HALO)   // 188
#define LDST  (LDW + 2)         // 190: even -> 8B-aligned v2f LDS reads, not bank-stride-64
#define NPART 2048

__device__ __forceinline__ int refl(int x, int n) {
    if (x < 0) x = -x;                 // jnp 'reflect': -1 -> 1
    if (x >= n) x = 2 * n - 2 - x;     // 512 -> 510
    return x;
}

__device__ __forceinline__ float wval(const float* wk, int t) {
    return (t >= 0 && t < WID) ? wk[t] : 0.0f;
}

__device__ __forceinline__ void make_weights(float* wk, int lane) {
    double s = 0.0;
    for (int t = 0; t < WID; ++t) { double d = (double)(t - PAD); s += exp(-0.5 * d * d); }
    for (int t = lane; t < WID; t += 32) {
        double d = (double)(t - PAD);
        wk[t] = (float)(exp(-0.5 * d * d) / s);
    }
}

// ---------------- Kernel 1: c = mean_ch(a - b), float4 vectorized ----------------
__global__ __launch_bounds__(256) void diff_kernel(const float* __restrict__ a,
                                                   const float* __restrict__ b,
                                                   float* __restrict__ c) {
    int idx = blockIdx.x * blockDim.x + threadIdx.x;   // over NIMG*HW/4
    if (idx >= NIMG * (HW / 4)) return;
    int img = idx >> 16;            // HW/4 = 65536
    int r   = idx & 65535;
    const float4* A = (const float4*)a + (size_t)img * 3 * 65536;
    const float4* B = (const float4*)b + (size_t)img * 3 * 65536;
    float4 a0 = A[r], a1 = A[r + 65536], a2 = A[r + 131072];
    float4 b0 = B[r], b1 = B[r + 65536], b2 = B[r + 131072];
    const float third = 1.0f / 3.0f;
    float4 o;
    o.x = (a0.x + a1.x + a2.x - b0.x - b1.x - b2.x) * third;
    o.y = (a0.y + a1.y + a2.y - b0.y - b1.y - b2.y) * third;
    o.z = (a0.z + a1.z + a2.z - b0.z - b1.z - b2.z) * third;
    o.w = (a0.w + a1.w + a2.w - b0.w - b1.w - b2.w) * third;
    ((float4*)c)[idx] = o;
}

// ------------- Kernel 2: horizontal 61-tap conv via WMMA f32 16x16x4 -------------
// out[m][x0+n] = sum_k tile[m][x0+k] * w[k-n]   (A = input rows, B = weight band)
__global__ __launch_bounds__(32) void hconv_kernel(const float* __restrict__ c,
                                                   float* __restrict__ h) {
    __shared__ __align__(16) float tile[16][LDST];
    __shared__ float wk[WID];
    int lane = threadIdx.x;
    make_weights(wk, lane);

    int bid  = blockIdx.x;          // 16 img * 32 bands * 4 chunks = 2048
    int xc   = bid & 3;
    int band = (bid >> 2) & 31;
    int img  = bid >> 7;
    int y0    = band * 16;
    int xbase = xc * XCHUNK;
    const float* cimg = c + (size_t)img * HW;

    for (int i = lane; i < 16 * LDW; i += 32) {        // 3008/32 = 94, uniform
        int r = i / LDW, cp = i % LDW;
        int x = refl(xbase + cp - PAD, W);
        tile[r][cp] = cimg[(y0 + r) * W + x];
    }
    __syncthreads();

    int n = lane & 15, half = lane >> 4;
    // Constant band matrix B[k][n] = w[k-n]; per step VGPR0 holds rows {4s,4s+2}, VGPR1 {4s+1,4s+3}
    v2f Bb[KSTEPS];
#pragma unroll
    for (int s = 0; s < KSTEPS; ++s) {
        int k0 = 4 * s + 2 * half;
        Bb[s].x = wval(wk, k0 - n);
        Bb[s].y = wval(wk, k0 + 1 - n);
    }

    float* himg = h + (size_t)img * HW;
    for (int tx = 0; tx < XCHUNK / 16; ++tx) {
        int x0 = tx * 16;
        v8f acc = {0.f, 0.f, 0.f, 0.f, 0.f, 0.f, 0.f, 0.f};
#pragma unroll
        for (int s = 0; s < KSTEPS; ++s) {
            int col = x0 + 4 * s + 2 * half;           // even -> 8B aligned
            v2f Am = *(const v2f*)&tile[n][col];
            acc = __builtin_amdgcn_wmma_f32_16x16x4_f32(false, Am, false, Bb[s],
                                                        (short)0, acc, false, false);
        }
#pragma unroll
        for (int v = 0; v < 8; ++v) {
            int row = v + 8 * half;
            himg[(y0 + row) * W + xbase + x0 + n] = acc[v];
        }
    }
}

// ------- Kernel 3: vertical conv via WMMA (A = weight band), square + reduce -------
// out[y0+m][x0+n] = sum_k w[k-m] * h[y0-30+k][x0+n]   (B = input columns)
__global__ __launch_bounds__(32) void vconv_kernel(const float* __restrict__ h,
                                                   float* __restrict__ partial) {
    __shared__ __align__(16) float tileV[16][LDST];    // [x][y], transposed staging
    __shared__ float wk[WID];
    int lane = threadIdx.x;
    make_weights(wk, lane);

    int bid = blockIdx.x;           // 16 img * 32 xtiles * 4 ychunks = 2048
    int yc  = bid & 3;
    int xt  = (bid >> 2) & 31;
    int img = bid >> 7;
    int ybase = yc * YCHUNK;
    int x0    = xt * 16;
    const float* himg = h + (size_t)img * HW;

    for (int i = lane; i < 16 * (YCHUNK + HALO); i += 32) {   // uniform
        int yy = i >> 4, xx = i & 15;
        int y = refl(ybase + yy - PAD, H);
        tileV[xx][yy] = himg[y * W + x0 + xx];
    }
    __syncthreads();

    int n = lane & 15, half = lane >> 4;
    // A band: A[m][k] = w[k-m]; lane's m = lane&15
    v2f Ab[KSTEPS];
#pragma unroll
    for (int s = 0; s < KSTEPS; ++s) {
        int k0 = 4 * s + 2 * half;
        Ab[s].x = wval(wk, k0 - n);
        Ab[s].y = wval(wk, k0 + 1 - n);
    }

    float sq = 0.0f;
    for (int ty = 0; ty < YCHUNK / 16; ++ty) {
        int yrel = ty * 16;
        v8f acc = {0.f, 0.f, 0.f, 0.f, 0.f, 0.f, 0.f, 0.f};
#pragma unroll
        for (int s = 0; s < KSTEPS; ++s) {
            int row = yrel + 4 * s + 2 * half;         // even -> 8B aligned
            v2f Bm = *(const v2f*)&tileV[n][row];
            acc = __builtin_amdgcn_wmma_f32_16x16x4_f32(false, Ab[s], false, Bm,
                                                        (short)0, acc, false, false);
        }
#pragma unroll
        for (int v = 0; v < 8; ++v) sq += acc[v] * acc[v];
    }
    // deterministic wave32 reduction
    for (int off = 16; off >= 1; off >>= 1) sq += __shfl_xor(sq, off, 32);
    if (lane == 0) partial[bid] = sq;
}

// ---------------- Kernel 4: deterministic final reduction ----------------
__global__ __launch_bounds__(256) void reduce_kernel(const float* __restrict__ partial,
                                                     float* __restrict__ out) {
    __shared__ float sm[256];
    int t = threadIdx.x;
    float v = 0.0f;
    for (int i = t; i < NPART; i += 256) v += partial[i];
    sm[t] = v;
    __syncthreads();
    for (int off = 128; off > 0; off >>= 1) {
        if (t < off) sm[t] += sm[t + off];
        __syncthreads();
    }
    if (t == 0) out[0] = sm[0] * (1.0f / (float)((size_t)NIMG * HW));
}

extern "C" void kernel_launch(void* const* d_in, const int* in_sizes, int n_in,
                              void* d_out, int out_size, void* d_ws, size_t ws_size,
                              hipStream_t stream) {
    (void)in_sizes; (void)n_in; (void)out_size; (void)ws_size;
    const float* a = (const float*)d_in[0];
    const float* b = (const float*)d_in[1];
    float* c       = (float*)d_ws;                       // 16 MB
    float* h       = c + (size_t)NIMG * HW;              // 16 MB
    float* partial = h + (size_t)NIMG * HW;              // 8 KB
    float* out     = (float*)d_out;

    diff_kernel<<<NIMG * (HW / 4) / 256, 256, 0, stream>>>(a, b, c);
    hconv_kernel<<<NPART, 32, 0, stream>>>(c, h);
    vconv_kernel<<<NPART, 32, 0, stream>>>(h, partial);
    reduce_kernel<<<1, 256, 0, stream>>>(partial, out);
}